// chess_attention_88596585382614
// MI455X (gfx1250) — compile-verified
//
#include <hip/hip_runtime.h>
#include <hip/hip_bf16.h>

// CDNA5 (gfx1250) fused chess-attention, fp32 WMMA 16x16x4 everywhere.
// HD=4 makes V_WMMA_F32_16X16X4_F32 a perfect shape match; problem is
// latency-bound (~2 GFLOP, ~17MB) so fp32 exactness costs nothing.
// Round 3: async global->LDS W1 stage with the correct builtin signature
// (global int4* src, lds int4* dst, imm offset, imm cpol).

typedef __attribute__((ext_vector_type(2))) float v2f;
typedef __attribute__((ext_vector_type(8))) float v8f;
typedef int v4i __attribute__((vector_size(16)));

#define WMMA4(A, B, C) \
  __builtin_amdgcn_wmma_f32_16x16x4_f32(false, (A), false, (B), (short)0, (C), false, false)

#if defined(__has_builtin)
#if __has_builtin(__builtin_amdgcn_global_load_async_to_lds_b128)
#define HAVE_ASYNC_LDS 1
#endif
#endif
#ifndef HAVE_ASYNC_LDS
#define HAVE_ASYNC_LDS 0
#endif

__device__ __forceinline__ void wait_asynccnt0() {
#if defined(__has_builtin)
#if __has_builtin(__builtin_amdgcn_s_wait_asynccnt)
  __builtin_amdgcn_s_wait_asynccnt(0);
  return;
#else
  asm volatile("s_wait_asynccnt 0x0" ::: "memory");
  return;
#endif
#else
  asm volatile("s_wait_asynccnt 0x0" ::: "memory");
#endif
}

static constexpr int Bn = 128, Dd = 32, Pp = 13, E = 16, Hh = 4, HD = 4;
static constexpr int L  = Dd * Pp;   // 416
static constexpr int M  = Bn * L;    // 53248 rows
static constexpr int MT = M / 16;    // 3328 16-row tiles

// ---------------------------------------------------------------------------
// Kernel 1: encoder.  enc = relu(board@W1a + p*w1[64] + d*w1[65] + b1) @ w2 + b2
// One wave per 16-row tile; W1 (64x64) staged in LDS (async on gfx1250);
// h bounced via LDS to re-fragment between the two GEMMs.
// ---------------------------------------------------------------------------
__global__ __launch_bounds__(128) void k_encoder(
    const float* __restrict__ board, const float* __restrict__ w1,
    const float* __restrict__ b1, const float* __restrict__ w2,
    const float* __restrict__ b2, float* __restrict__ enc) {
  __shared__ float sW1[64 * 64];      // 16 KB
  __shared__ float sH[4][16 * 64];    // 16 KB (h tiles, one per wave)

  const int tid = threadIdx.x;
#if HAVE_ASYNC_LDS
  // 4096 floats = 128 threads x 8 async B128 copies; overlaps with the
  // per-lane A-fragment preload below (ASYNCcnt-tracked, no VGPR transit).
#pragma unroll
  for (int j = 0; j < 8; ++j) {
    const int c = (tid + 128 * j) * 4;
    __builtin_amdgcn_global_load_async_to_lds_b128(
        (__attribute__((address_space(1))) v4i*)(w1 + c),
        (__attribute__((address_space(3))) v4i*)(sW1 + c),
        0, 0);
  }
#else
  for (int i = tid; i < 64 * 64; i += 128) sW1[i] = w1[i];
#endif

  const int wave = tid >> 5, lane = tid & 31;
  const int tile = blockIdx.x * 4 + wave;
  const int row = lane & 15, col = lane & 15;
  const int k0 = (lane >> 4) << 1;        // 0 or 2 (K position in fragment)
  const int rowbase = (lane >> 4) << 3;   // 0 or 8 (C/D row group)

  // Preload A (16x64 board rows) as 16 K-step fragments (overlaps async stage).
  const float* arow = board + (size_t)(tile * 16 + row) * 64;
  v2f af[16];
#pragma unroll
  for (int s = 0; s < 16; ++s) {
    af[s].x = arow[4 * s + k0];
    af[s].y = arow[4 * s + k0 + 1];
  }

#if HAVE_ASYNC_LDS
  wait_asynccnt0();
#endif
  __syncthreads();

  // GEMM1: h(16x64) = A(16x64) @ W1a(64x64), 4 N-tiles x 16 K-steps.
#pragma unroll
  for (int nt = 0; nt < 4; ++nt) {
    v8f acc = {};
#pragma unroll
    for (int s = 0; s < 16; ++s) {
      v2f bf;
      bf.x = sW1[(4 * s + k0) * 64 + nt * 16 + col];
      bf.y = sW1[(4 * s + k0 + 1) * 64 + nt * 16 + col];
      acc = WMMA4(af[s], bf, acc);
    }
    const int n = nt * 16 + col;
    const float w1p = w1[64 * 64 + n];   // piece-id weight row
    const float w1d = w1[65 * 64 + n];   // depth-id weight row
    const float b1n = b1[n];
#pragma unroll
    for (int r = 0; r < 8; ++r) {
      int m = tile * 16 + rowbase + r;
      int l = m % L;
      int dd = l / 13;
      int pp = l - dd * 13;
      float hv = acc[r] + (float)pp * w1p + (float)dd * w1d + b1n;
      sH[wave][(rowbase + r) * 64 + n] = hv > 0.f ? hv : 0.f;
    }
  }
  __syncthreads();   // re-fragment h (cross-lane through LDS)

  // GEMM2: enc(16x16) = h(16x64) @ w2(64x16) + b2.
  v8f acc = {};
#pragma unroll
  for (int s = 0; s < 16; ++s) {
    v2f a, bf;
    a.x = sH[wave][row * 64 + 4 * s + k0];
    a.y = sH[wave][row * 64 + 4 * s + k0 + 1];
    bf.x = w2[(4 * s + k0) * 16 + col];
    bf.y = w2[(4 * s + k0 + 1) * 16 + col];
    acc = WMMA4(a, bf, acc);
  }
  const float b2n = b2[col];
#pragma unroll
  for (int r = 0; r < 8; ++r)
    enc[(size_t)(tile * 16 + rowbase + r) * 16 + col] = acc[r] + b2n;
}

// ---------------------------------------------------------------------------
// Kernel 2: qkv = enc @ w_qkv^T + b_qkv, scattered to (b,h,l,hd) q/k/v bufs.
// ---------------------------------------------------------------------------
__global__ __launch_bounds__(128) void k_qkv(
    const float* __restrict__ enc, const float* __restrict__ wqkv,
    const float* __restrict__ bqkv, float* __restrict__ qb,
    float* __restrict__ kb, float* __restrict__ vb) {
  const int tid = threadIdx.x, wave = tid >> 5, lane = tid & 31;
  const int tile = blockIdx.x * 4 + wave;
  const int row = lane & 15, col = lane & 15;
  const int k0 = (lane >> 4) << 1, rowbase = (lane >> 4) << 3;

  v2f af[4];
#pragma unroll
  for (int s = 0; s < 4; ++s) {
    af[s].x = enc[(size_t)(tile * 16 + row) * 16 + 4 * s + k0];
    af[s].y = enc[(size_t)(tile * 16 + row) * 16 + 4 * s + k0 + 1];
  }
#pragma unroll
  for (int nt = 0; nt < 3; ++nt) {     // 0:q 1:k 2:v
    v8f acc = {};
#pragma unroll
    for (int s = 0; s < 4; ++s) {
      v2f bf;                          // B[k][j] = wqkv[j][k]
      bf.x = wqkv[(nt * 16 + col) * 16 + 4 * s + k0];
      bf.y = wqkv[(nt * 16 + col) * 16 + 4 * s + k0 + 1];
      acc = WMMA4(af[s], bf, acc);
    }
    const float bias = bqkv[nt * 16 + col];
    float* dst = (nt == 0) ? qb : (nt == 1) ? kb : vb;
    const int head = col >> 2, hd = col & 3;
#pragma unroll
    for (int r = 0; r < 8; ++r) {
      int m = tile * 16 + rowbase + r;
      int bb = m / L, l = m - bb * L;
      dst[(((size_t)bb * Hh + head) * L + l) * HD + hd] = acc[r] + bias;
    }
  }
}

// ---------------------------------------------------------------------------
// Kernel 3: attention. One wave per (b,h,16-query tile). Scores via single
// K=4 WMMA per 16-col tile; exact softmax over a 16x416 LDS strip; p@v via
// chained K=4 WMMAs with v zero-padded to N=16.
// ---------------------------------------------------------------------------
__global__ __launch_bounds__(128) void k_attn(
    const float* __restrict__ qb_, const float* __restrict__ kb_,
    const float* __restrict__ vb_, float* __restrict__ ao) {
  __shared__ float sS[4][16 * L];   // 104 KB: unnormalized probs per wave
  __shared__ float sSum[4][16];

  const int tid = threadIdx.x, wave = tid >> 5, lane = tid & 31;
  const int u = blockIdx.x * 4 + wave;     // < 128*4*26
  const int qt = u % 26;
  const int bh = u / 26;
  const int b = bh >> 2, h = bh & 3;
  const float* qb = qb_ + (size_t)bh * L * HD;
  const float* kb = kb_ + (size_t)bh * L * HD;
  const float* vb = vb_ + (size_t)bh * L * HD;
  const int row = lane & 15, col = lane & 15;
  const int k0 = (lane >> 4) << 1, rowbase = (lane >> 4) << 3;

  v2f aq;   // q fragment reused for all 26 K tiles
  aq.x = qb[(qt * 16 + row) * HD + k0];
  aq.y = qb[(qt * 16 + row) * HD + k0 + 1];

  for (int kt = 0; kt < 26; ++kt) {
    v2f bk;                                // B[d][j] = k[j][d]
    bk.x = kb[(kt * 16 + col) * HD + k0];
    bk.y = kb[(kt * 16 + col) * HD + k0 + 1];
    v8f s8 = {};
    s8 = WMMA4(aq, bk, s8);
#pragma unroll
    for (int r = 0; r < 8; ++r)
      sS[wave][(rowbase + r) * L + kt * 16 + col] = s8[r] * 0.5f;  // 1/sqrt(4)
  }
  __syncthreads();

  // Softmax: 16 rows, 416 = 13*32 cols; whole wave per row.
  for (int r = 0; r < 16; ++r) {
    float vals[13];
    float mx = -3.4e38f;
#pragma unroll
    for (int i = 0; i < 13; ++i) {
      vals[i] = sS[wave][r * L + lane + i * 32];
      mx = fmaxf(mx, vals[i]);
    }
#pragma unroll
    for (int off = 16; off > 0; off >>= 1) mx = fmaxf(mx, __shfl_xor(mx, off, 32));
    float sum = 0.f;
#pragma unroll
    for (int i = 0; i < 13; ++i) {
      float e = __expf(vals[i] - mx);
      sum += e;
      sS[wave][r * L + lane + i * 32] = e;
    }
#pragma unroll
    for (int off = 16; off > 0; off >>= 1) sum += __shfl_xor(sum, off, 32);
    if (lane == 0) sSum[wave][r] = sum;
  }
  __syncthreads();

  // out(16x4) = p(16x416) @ v(416x4): 26*4 chained K=4 WMMAs, v padded N=16.
  v8f o = {};
  for (int kt = 0; kt < 26; ++kt) {
#pragma unroll
    for (int s2 = 0; s2 < 4; ++s2) {
      const int ks = kt * 16 + s2 * 4;
      v2f a, bv;
      a.x = sS[wave][row * L + ks + k0];
      a.y = sS[wave][row * L + ks + k0 + 1];
      bv.x = (col < HD) ? vb[(ks + k0) * HD + col] : 0.f;
      bv.y = (col < HD) ? vb[(ks + k0 + 1) * HD + col] : 0.f;
      o = WMMA4(a, bv, o);
    }
  }
  if (col < HD) {
#pragma unroll
    for (int r = 0; r < 8; ++r) {
      float inv = 1.0f / sSum[wave][rowbase + r];
      ao[((size_t)b * L + qt * 16 + rowbase + r) * E + h * HD + col] = o[r] * inv;
    }
  }
}

// ---------------------------------------------------------------------------
// Kernel 4: out = ao @ w_out^T + b_out  (16x16, 4 K-steps)
// ---------------------------------------------------------------------------
__global__ __launch_bounds__(128) void k_out(
    const float* __restrict__ ao, const float* __restrict__ wout,
    const float* __restrict__ bout, float* __restrict__ out) {
  const int tid = threadIdx.x, wave = tid >> 5, lane = tid & 31;
  const int tile = blockIdx.x * 4 + wave;
  const int row = lane & 15, col = lane & 15;
  const int k0 = (lane >> 4) << 1, rowbase = (lane >> 4) << 3;

  v8f acc = {};
#pragma unroll
  for (int s = 0; s < 4; ++s) {
    v2f a, bf;
    a.x = ao[(size_t)(tile * 16 + row) * 16 + 4 * s + k0];
    a.y = ao[(size_t)(tile * 16 + row) * 16 + 4 * s + k0 + 1];
    bf.x = wout[col * 16 + 4 * s + k0];     // B[k][n] = w_out[n][k]
    bf.y = wout[col * 16 + 4 * s + k0 + 1];
    acc = WMMA4(a, bf, acc);
  }
  const float bias = bout[col];
#pragma unroll
  for (int r = 0; r < 8; ++r)
    out[(size_t)(tile * 16 + rowbase + r) * 16 + col] = acc[r] + bias;
}

// ---------------------------------------------------------------------------
extern "C" void kernel_launch(void* const* d_in, const int* in_sizes, int n_in,
                              void* d_out, int out_size, void* d_ws, size_t ws_size,
                              hipStream_t stream) {
  const float* board = (const float*)d_in[0];
  const float* w1    = (const float*)d_in[1];
  const float* b1    = (const float*)d_in[2];
  const float* w2    = (const float*)d_in[3];
  const float* b2    = (const float*)d_in[4];
  const float* wqkv  = (const float*)d_in[5];
  const float* bqkv  = (const float*)d_in[6];
  const float* wout  = (const float*)d_in[7];
  const float* bout  = (const float*)d_in[8];
  float* out = (float*)d_out;

  float* ws  = (float*)d_ws;               // 5 buffers of M*E floats (~17 MB)
  float* enc = ws;
  float* qb  = enc + (size_t)M * E;
  float* kb  = qb + (size_t)M * E;
  float* vb  = kb + (size_t)M * E;
  float* ao  = vb + (size_t)M * E;

  dim3 blk(128);
  k_encoder<<<MT / 4, blk, 0, stream>>>(board, w1, b1, w2, b2, enc);
  k_qkv<<<MT / 4, blk, 0, stream>>>(enc, wqkv, bqkv, qb, kb, vb);
  k_attn<<<(Bn * Hh * 26) / 4, blk, 0, stream>>>(qb, kb, vb, ao);
  k_out<<<MT / 4, blk, 0, stream>>>(ao, wout, bout, out);
}